// AttentionLayerCrossSegments_37014028157214
// MI455X (gfx1250) — compile-verified
//
#include <hip/hip_runtime.h>

// ---------------- problem constants (match reference) ----------------
#define BB   4
#define PP   1024
#define DD   512
#define HH   8
#define DKK  64
#define BAND 32        // SPARSITY/2
#define WIN  80        // band window per 16-row block: [l0-32, l0+47]
#define LDSS 84        // padded LDS row stride (floats) -> conflict-free

typedef float v2f __attribute__((ext_vector_type(2)));
typedef float v8f __attribute__((ext_vector_type(8)));

// D = A(16x4 f32) * B(4x16 f32) + C(16x16 f32), full fp32 precision.
static __device__ __forceinline__ v8f wmma4(v2f a, v2f b, v8f c) {
    return __builtin_amdgcn_wmma_f32_16x16x4_f32(
        /*neg_a=*/false, a, /*neg_b=*/false, b,
        /*c_mod=*/(short)0, c, /*reuse_a=*/false, /*reuse_b=*/false);
}

// ---------------------------------------------------------------------
// C[M x 512] = A[M x 512] @ W[512 x 512] + bias[512]
// block = 4 waves; wave w computes the 16x16 tile at (m0, n0 + 16*w).
// A-fragment lane map (ISA 32-bit A 16x4): M = lane&15, K = 2*(lane>>4)+{0,1}
// B-fragment lane map (mirror):            N = lane&15, K = 2*(lane>>4)+{0,1}
// C/D lane map: row = r + 8*(lane>>4), col = lane&15  (r = vgpr 0..7)
// ---------------------------------------------------------------------
__global__ __launch_bounds__(128) void gemm_bias_f32(
    const float* __restrict__ A, const float* __restrict__ W,
    const float* __restrict__ bias, float* __restrict__ C)
{
    const int lane = threadIdx.x & 31;
    const int wave = threadIdx.x >> 5;
    const int n16  = lane & 15;
    const int kh   = lane >> 4;                 // 0/1 -> K offset 2*kh
    const int m0   = blockIdx.x * 16;
    const int n0   = blockIdx.y * 64 + wave * 16;

    const float* arow = A + (size_t)(m0 + n16) * DD;   // this lane's A row
    const float* bcol = W + n0 + n16;                  // this lane's W column

    v8f acc = {};
    for (int k0 = 0; k0 < DD; k0 += 4) {
        const int ka = k0 + 2 * kh;
        v2f a = *(const v2f*)(arow + ka);              // A[m][ka], A[m][ka+1]
        v2f b;
        b.x = bcol[(size_t)ka * DD];                   // W[ka  ][n]
        b.y = bcol[(size_t)(ka + 1) * DD];             // W[ka+1][n]
        acc = wmma4(a, b, acc);
    }

    const float bn = bias[n0 + n16];
    #pragma unroll
    for (int r = 0; r < 8; ++r) {
        const int m = m0 + r + 8 * kh;
        C[(size_t)m * DD + n0 + n16] = acc[r] + bn;
    }
}

// ---------------------------------------------------------------------
// Banded attention for one 16-row block of one (b, h):
//   scores (16 x 80) via WMMA  ->  scaled+masked to LDS
//   row softmax (lanes 0..15)  ->  probs to LDS + band write to Aout
//   O (16 x 64) = probs @ V via WMMA -> concat buffer
// Zeros outside the band in Aout are provided by a prior memset.
// ---------------------------------------------------------------------
__global__ __launch_bounds__(32) void band_attn(
    const float* __restrict__ Q, const float* __restrict__ K,
    const float* __restrict__ V, float* __restrict__ concat,
    float* __restrict__ Aout, int row_off)
{
    __shared__ float sS[16 * LDSS];

    const int lane = threadIdx.x;
    const int n16  = lane & 15;
    const int kh   = lane >> 4;
    const int l0   = blockIdx.x * 16;
    const int h    = blockIdx.y;
    const int b    = blockIdx.z;
    const int s_base = l0 - BAND;                      // window start (may be <0)

    const float* qrow = Q + ((size_t)b * PP + (l0 + n16)) * DD + h * DKK;

    // ---- scores: 5 tiles of 16x16 over the 80-column window ----
    for (int t = 0; t < 5; ++t) {
        int sg = s_base + t * 16 + n16;                // this lane's key index
        int sc = sg < 0 ? 0 : (sg >= PP ? PP - 1 : sg);
        const float* krow = K + ((size_t)b * PP + sc) * DD + h * DKK;

        v8f c = {};
        #pragma unroll
        for (int k0 = 0; k0 < DKK; k0 += 4) {
            v2f a  = *(const v2f*)(qrow + k0 + 2 * kh); // Q[l][e], Q[l][e+1]
            v2f bv = *(const v2f*)(krow + k0 + 2 * kh); // K[s][e], K[s][e+1] (B = K^T)
            c = wmma4(a, bv, c);
        }
        #pragma unroll
        for (int r = 0; r < 8; ++r) {
            const int m = r + 8 * kh;                  // row within tile
            const int l = l0 + m;
            const int s = s_base + t * 16 + n16;
            const bool ok = (s >= 0) && (s < PP) &&
                            (s - l <= BAND) && (l - s <= BAND);
            sS[m * LDSS + t * 16 + n16] = ok ? 0.125f * c[r] : -__builtin_inff();
        }
    }
    __syncthreads();

    // ---- row softmax + write band probabilities (lanes 0..15, one row each) ----
    if (lane < 16) {
        float* row = sS + lane * LDSS;
        float mx = -__builtin_inff();
        for (int j = 0; j < WIN; ++j) mx = fmaxf(mx, row[j]);
        float sum = 0.f;
        for (int j = 0; j < WIN; ++j) {
            float e = __expf(row[j] - mx);             // exp(-inf)=0 for masked
            row[j] = e;
            sum += e;
        }
        const float inv = 1.0f / sum;                  // diagonal always in band -> sum>0
        float* arow = Aout + ((size_t)(b * HH + h) * PP + (l0 + lane)) * PP;
        for (int j = 0; j < WIN; ++j) {
            float p = row[j] * inv;
            row[j] = p;
            int s = s_base + j;
            if (s >= 0 && s < PP) arow[s] = p;
        }
    }
    __syncthreads();

    // ---- O = probs(16x80) @ V(80x64), 4 output tiles of 16x16 ----
    const size_t obase = ((size_t)b * 2 * PP + row_off + l0) * DD + h * DKK;
    #pragma unroll
    for (int nt = 0; nt < 4; ++nt) {
        const int d0 = nt * 16;
        v8f c = {};
        for (int kk = 0; kk < WIN; kk += 4) {
            const int ka = kk + 2 * kh;
            v2f a;                                     // A-frag: M=n16, K=ka..ka+1
            a.x = sS[n16 * LDSS + ka];
            a.y = sS[n16 * LDSS + ka + 1];
            int s0 = s_base + ka, s1 = s0 + 1;
            int c0 = s0 < 0 ? 0 : (s0 >= PP ? PP - 1 : s0);
            int c1 = s1 < 0 ? 0 : (s1 >= PP ? PP - 1 : s1);
            v2f bv;                                    // B-frag: N=n16, K rows of V
            bv.x = V[((size_t)b * PP + c0) * DD + h * DKK + d0 + n16];
            bv.y = V[((size_t)b * PP + c1) * DD + h * DKK + d0 + n16];
            c = wmma4(a, bv, c);                       // prob==0 kills clamped rows
        }
        #pragma unroll
        for (int r = 0; r < 8; ++r) {
            concat[obase + (size_t)(r + 8 * kh) * DD + d0 + n16] = c[r];
        }
    }
}

// ---------------------------------------------------------------------
extern "C" void kernel_launch(void* const* d_in, const int* in_sizes, int n_in,
                              void* d_out, int out_size, void* d_ws, size_t ws_size,
                              hipStream_t stream) {
    (void)in_sizes; (void)n_in; (void)out_size; (void)ws_size;

    const float* queries = (const float*)d_in[0];
    const float* keys    = (const float*)d_in[1];
    const float* values  = (const float*)d_in[2];
    const float* Wq = (const float*)d_in[3];
    const float* bq = (const float*)d_in[4];
    const float* Wk = (const float*)d_in[5];
    const float* bk = (const float*)d_in[6];
    const float* Wv = (const float*)d_in[7];
    const float* bv = (const float*)d_in[8];
    const float* Wo = (const float*)d_in[9];
    const float* bo = (const float*)d_in[10];

    float* out = (float*)d_out;                        // (B, 2P, D)
    float* A1  = out + (size_t)BB * 2 * PP * DD;       // (B, H, P, P)
    float* A2  = A1 + (size_t)BB * HH * PP * PP;       // (B, H, P, P)

    float* Qp = (float*)d_ws;                          // (B*P, 512)
    float* Kp = Qp + (size_t)BB * PP * DD;
    float* Vp = Kp + (size_t)BB * PP * DD;
    float* Cc = Vp + (size_t)BB * PP * DD;             // concat (B, 2P, 512)

    // Zero A1+A2 (adjacent): the band kernels overwrite only |i-j|<=32.
    hipMemsetAsync(A1, 0, (size_t)2 * BB * HH * PP * PP * sizeof(float), stream);

    // Q/K/V projections
    dim3 gp(BB * PP / 16, DD / 64);
    gemm_bias_f32<<<gp, 128, 0, stream>>>(queries, Wq, bq, Qp);
    gemm_bias_f32<<<gp, 128, 0, stream>>>(keys,    Wk, bk, Kp);
    gemm_bias_f32<<<gp, 128, 0, stream>>>(values,  Wv, bv, Vp);

    // Banded attentions: attn(q,k,v) -> out1/A1 ; attn(k,q,q) -> out2/A2
    dim3 ga(PP / 16, HH, BB);
    band_attn<<<ga, 32, 0, stream>>>(Qp, Kp, Vp, Cc, A1, 0);
    band_attn<<<ga, 32, 0, stream>>>(Kp, Qp, Qp, Cc, A2, PP);

    // Output projection on the concatenated buffer
    dim3 go(BB * 2 * PP / 16, DD / 64);
    gemm_bias_f32<<<go, 128, 0, stream>>>(Cc, Wo, bo, out);
}